// Conv2d_6459630814025
// MI455X (gfx1250) — compile-verified
//
#include <hip/hip_runtime.h>
#include <stdint.h>

// Problem constants (match reference)
#define KS   3
#define B_   8
#define C_   32
#define O_   64
#define H_   48
#define W_   48
#define K_   (C_ * KS * KS)        // 288
#define WP_  (W_ + 2)              // 50 (padded slab width)

#define SW_ELEMS  (O_ * K_)        // 18432 floats = 72 KB
#define SIN_ELEMS (C_ * KS * WP_)  // 4800 floats  = 18.75 KB

typedef unsigned int u32x4 __attribute__((ext_vector_type(4)));
typedef int          i32x8 __attribute__((ext_vector_type(8)));

// ---- TDM: async DMA of one contiguous 18432-f32 block into LDS offset 0 ----
// 2-SGPR-group form (VADDR2/3 = NULL) for <=2D tensors; 1 row x 18432 elems.
__device__ __forceinline__ void tdm_load_72kb(const void* gsrc) {
    uint64_t ga = (uint64_t)(uintptr_t)gsrc;
    u32x4 g0;
    g0[0] = 1u;                                     // count=1, user desc
    g0[1] = 0u;                                     // lds_addr = 0
    g0[2] = (uint32_t)(ga & 0xFFFFFFFFull);         // global_addr[31:0]
    g0[3] = (uint32_t)((ga >> 32) & 0x01FFFFFFull)  // global_addr[56:32]
          | (2u << 30);                             // type = 2 (image)
    i32x8 g1;
    g1[0] = 0x00020000;                 // workgroup_mask=0, data_size=2 (4B)
    g1[1] = (int)(((unsigned)(SW_ELEMS & 0xFFFF)) << 16); // tensor_dim0 lo16
    g1[2] = (int)(1u << 16);            // tensor_dim0 hi16=0 | tensor_dim1=1
    g1[3] = (int)(((unsigned)(SW_ELEMS & 0xFFFF)) << 16); // tile_dim0=18432
    g1[4] = 1;                          // tile_dim1=1, tile_dim2=0
    g1[5] = SW_ELEMS;                   // tensor_dim0_stride[31:0]
    g1[6] = 0;                          // stride0 hi | stride1 lo
    g1[7] = 0;                          // stride1 hi
    asm volatile("tensor_load_to_lds %0, %1"
                 :
                 : "s"(g0), "s"(g1)
                 : "memory");
}

// ---- Pre-pass: transpose weights (O,K) -> (K,O) into global scratch ----
__global__ __launch_bounds__(256)
void transpose_w_kernel(const float* __restrict__ w, float* __restrict__ wt) {
    int t = blockIdx.x * 256 + threadIdx.x;   // 0..18431, coalesced writes
    int k = t >> 6;                           // t / 64
    int o = t & 63;                           // t % 64
    wt[t] = w[o * K_ + k];
}

// ---- Main kernel: transposed weights, one ds_load_b128 per tap ----
__global__ __launch_bounds__(256)
void cheb_conv_tw_kernel(const float* __restrict__ in,
                         const float* __restrict__ wt,   // (K_, O_) in d_ws
                         float* __restrict__ out) {
    __shared__ __align__(16) float smem[SW_ELEMS + SIN_ELEMS];
    float* s_wt = smem;             // [K_][O_]   weights, k-major (transposed)
    float* s_in = smem + SW_ELEMS;  // [C_][3][WP_] padded 3-row input slab

    const int tid = threadIdx.x;
    const int y   = blockIdx.x;     // output row 0..47
    const int b   = blockIdx.y;     // batch 0..7

    if (tid < 32) tdm_load_72kb(wt);   // wave-uniform gate: wave 0 issues DMA

    // Cooperative load of the zero-padded input slab (overlaps the TDM DMA).
    for (int idx = tid; idx < SIN_ELEMS; idx += 256) {
        int c  = idx / (KS * WP_);
        int r  = idx % (KS * WP_);
        int kh = r / WP_;
        int xx = (r % WP_) - 1;          // -1..48
        int yy = y + kh - 1;
        float v = 0.0f;
        if (xx >= 0 && xx < W_ && yy >= 0 && yy < H_)
            v = in[((b * C_ + c) * H_ + yy) * W_ + xx];
        s_in[idx] = v;
    }

    if (tid < 32) __builtin_amdgcn_s_wait_tensorcnt(0);
    __syncthreads();

    // Each thread owns 4 output channels at one x (768 pairs, 3 per thread).
    // Per tap: 1 input ds_load_b32 + 1 weight ds_load_b128 + 4 v_sub + 4 v_max.
    for (int p = tid; p < W_ * (O_ / 4); p += 256) {
        int x  = p % W_;
        int ob = (p / W_) << 2;                 // o base: 0,4,...,60
        float a0 = 0.0f, a1 = 0.0f, a2 = 0.0f, a3 = 0.0f;

        for (int c = 0; c < C_; ++c) {
            const float* ip = s_in + (c * KS) * WP_ + x;
#pragma unroll
            for (int kh = 0; kh < KS; ++kh) {
#pragma unroll
                for (int kw = 0; kw < KS; ++kw) {
                    float  v  = ip[kh * WP_ + kw];
                    int    k  = (c * KS + kh) * KS + kw;
                    float4 w4 = *(const float4*)(s_wt + (k << 6) + ob);
                    a0 = fmaxf(a0, fabsf(v - w4.x));
                    a1 = fmaxf(a1, fabsf(v - w4.y));
                    a2 = fmaxf(a2, fabsf(v - w4.z));
                    a3 = fmaxf(a3, fabsf(v - w4.w));
                }
            }
        }

        int ob0 = ((b * O_ + ob) * H_ + y) * W_ + x;
        out[ob0]             = a0;
        out[ob0 + H_ * W_]   = a1;
        out[ob0 + 2*H_*W_]   = a2;
        out[ob0 + 3*H_*W_]   = a3;
    }
}

// ---- Fallback (no scratch): untransposed weights, scalar LDS weight reads ----
__global__ __launch_bounds__(256)
void cheb_conv_kernel(const float* __restrict__ in,
                      const float* __restrict__ wgt,
                      float* __restrict__ out) {
    __shared__ __align__(16) float smem[SW_ELEMS + SIN_ELEMS];
    float* s_w  = smem;             // [O_][K_]
    float* s_in = smem + SW_ELEMS;

    const int tid = threadIdx.x;
    const int y   = blockIdx.x;
    const int b   = blockIdx.y;

    if (tid < 32) tdm_load_72kb(wgt);

    for (int idx = tid; idx < SIN_ELEMS; idx += 256) {
        int c  = idx / (KS * WP_);
        int r  = idx % (KS * WP_);
        int kh = r / WP_;
        int xx = (r % WP_) - 1;
        int yy = y + kh - 1;
        float v = 0.0f;
        if (xx >= 0 && xx < W_ && yy >= 0 && yy < H_)
            v = in[((b * C_ + c) * H_ + yy) * W_ + xx];
        s_in[idx] = v;
    }

    if (tid < 32) __builtin_amdgcn_s_wait_tensorcnt(0);
    __syncthreads();

    for (int p = tid; p < W_ * (O_ / 4); p += 256) {
        int x  = p % W_;
        int ob = (p / W_) << 2;
        const float* w0 = s_w + ob * K_;
        float a0 = 0.0f, a1 = 0.0f, a2 = 0.0f, a3 = 0.0f;

        for (int c = 0; c < C_; ++c) {
            const float* ip = s_in + (c * KS) * WP_ + x;
            const float* wp = w0 + c * (KS * KS);
#pragma unroll
            for (int kh = 0; kh < KS; ++kh) {
#pragma unroll
                for (int kw = 0; kw < KS; ++kw) {
                    float v = ip[kh * WP_ + kw];
                    int   k = kh * KS + kw;
                    float d;
                    d = v - wp[k];          a0 = fmaxf(a0, fabsf(d));
                    d = v - wp[k + K_];     a1 = fmaxf(a1, fabsf(d));
                    d = v - wp[k + 2*K_];   a2 = fmaxf(a2, fabsf(d));
                    d = v - wp[k + 3*K_];   a3 = fmaxf(a3, fabsf(d));
                }
            }
        }

        int ob0 = ((b * O_ + ob) * H_ + y) * W_ + x;
        out[ob0]             = a0;
        out[ob0 + H_ * W_]   = a1;
        out[ob0 + 2*H_*W_]   = a2;
        out[ob0 + 3*H_*W_]   = a3;
    }
}

extern "C" void kernel_launch(void* const* d_in, const int* in_sizes, int n_in,
                              void* d_out, int out_size, void* d_ws, size_t ws_size,
                              hipStream_t stream) {
    const float* in  = (const float*)d_in[0];   // (8,32,48,48) f32
    const float* wgt = (const float*)d_in[1];   // (64,32,3,3)  f32
    float* out = (float*)d_out;                 // (8,64,48,48) f32
    dim3 grid(H_, B_);                          // 48 x 8 = 384 blocks

    if (ws_size >= (size_t)SW_ELEMS * sizeof(float)) {
        float* wt = (float*)d_ws;
        transpose_w_kernel<<<dim3(SW_ELEMS / 256), 256, 0, stream>>>(wgt, wt);
        cheb_conv_tw_kernel<<<grid, 256, 0, stream>>>(in, wt, out);
    } else {
        cheb_conv_kernel<<<grid, 256, 0, stream>>>(in, wgt, out);
    }
}